// _Core_76897094467842
// MI455X (gfx1250) — compile-verified
//
#include <hip/hip_runtime.h>
#include <cstdint>

#define TPB 256
#define KK  32

// ---------------------------------------------------------------------------
// Pack gaussians into 64B AoS records: 3 float4 per gaussian (stride 4 float4)
//   rec[0] = {mean.x, mean.y, a, b}
//   rec[1] = {c, opacity, tau, 0}
//   rec[2] = {col.r, col.g, col.b, 0}
// Gather cost per composite drops from 6 loads over 5 arrays to 3x B128
// within one 64B-aligned record (<=2 cache lines, usually 1).
// ---------------------------------------------------------------------------
__global__ __launch_bounds__(TPB) void pack_gauss(
    const float* __restrict__ means, const float* __restrict__ conics,
    const float* __restrict__ colors, const float* __restrict__ opac,
    const float* __restrict__ taus, float4* __restrict__ packed, int N)
{
    int g = blockIdx.x * TPB + threadIdx.x;
    if (g >= N) return;
    float4 p0 = make_float4(means[2*g], means[2*g+1], conics[3*g], conics[3*g+1]);
    float4 p1 = make_float4(conics[3*g+2], opac[g], taus[g], 0.f);
    float4 p2 = make_float4(colors[3*g], colors[3*g+1], colors[3*g+2], 0.f);
    packed[4*(size_t)g + 0] = p0;
    packed[4*(size_t)g + 1] = p1;
    packed[4*(size_t)g + 2] = p2;
}

// ---------------------------------------------------------------------------
// Fast path: K == 32, one thread per pixel. Each wave DMA-stages its 4KB
// slice of work_g into LDS with CDNA5 GLOBAL_LOAD_ASYNC_TO_LDS_B128
// (ASYNCcnt-tracked, no VGPR round-trip), overlapping the uniform loads and
// pixel-coordinate math with the DMA before s_wait_asynccnt.
// ---------------------------------------------------------------------------
__global__ __launch_bounds__(TPB) void render_k32(
    const float4* __restrict__ packed, const int* __restrict__ work_g,
    const float* __restrict__ background, const int* __restrict__ wptr,
    float* __restrict__ out)
{
    __shared__ int sIdx[TPB * KK];          // 32 KB: indices for this block's 256 pixels

    const int t    = threadIdx.x;
    const int p    = blockIdx.x * TPB + t;
    const int lane = t & 31;
    // wave id, made provably wave-uniform for the SGPR asm operand
    const int wave = __builtin_amdgcn_readfirstlane(t >> 5);

    // This wave stages 1024 ints (32 pixels x 32 indices) = 4KB:
    //  global  [work_g + (block*256 + wave*32)*32 , +1024)
    //  LDS     [sIdx   + wave*1024               , +1024)
    const int* gsrc = work_g + ((size_t)blockIdx.x * TPB + (size_t)wave * 32) * KK;
    unsigned ldsBase = (unsigned)(uintptr_t)(&sIdx[wave * 1024]);
#pragma unroll
    for (int j = 0; j < 8; ++j) {
        unsigned off = (unsigned)((j * 32 + lane) * 16);   // 16B per lane per step
        unsigned lds = ldsBase + off;
        asm volatile("global_load_async_to_lds_b128 %0, %1, %2"
                     :: "v"(lds), "v"(off), "s"(gsrc)
                     : "memory");
    }

    // Overlap with the DMA: uniform loads + coordinate math
    const int   W  = wptr[0];
    const float bg0 = background[0];
    const float bg1 = background[1];
    const float bg2 = background[2];
    const float x  = (float)(p % W) + 0.5f;
    const float y  = (float)(p / W) + 0.5f;

    asm volatile("s_wait_asynccnt 0" ::: "memory");  // wave reads only its own slice

    float T = 1.f, cr = 0.f, cg = 0.f, cb = 0.f;
    const int* myIdx = &sIdx[t * KK];

#pragma unroll 4
    for (int k = 0; k < KK; ++k) {
        int gi = myIdx[k];
        float4 p0 = packed[4*(size_t)gi + 0];
        float4 p1 = packed[4*(size_t)gi + 1];
        float4 p2 = packed[4*(size_t)gi + 2];
        float dx = x - p0.x;
        float dy = y - p0.y;
        float q  = fmaf(p0.z * dx, dx, fmaf(2.f * p0.w * dx, dy, p1.x * dy * dy));
        float al = (q <= p1.z) ? p1.y * __expf(-0.5f * q) : 0.f;
        al = fminf(al, 0.99f);
        float w = al * T;
        cr = fmaf(w, p2.x, cr);
        cg = fmaf(w, p2.y, cg);
        cb = fmaf(w, p2.z, cb);
        T *= fmaxf(1.f - al, 1e-8f);
    }

    out[3*(size_t)p + 0] = fmaf(T, bg0, cr);
    out[3*(size_t)p + 1] = fmaf(T, bg1, cg);
    out[3*(size_t)p + 2] = fmaf(T, bg2, cb);
}

// ---------------------------------------------------------------------------
// Generic fallback (any K, any P, no workspace requirement)
// ---------------------------------------------------------------------------
__global__ __launch_bounds__(TPB) void render_generic(
    const float* __restrict__ means, const float* __restrict__ conics,
    const float* __restrict__ colors, const float* __restrict__ opac,
    const float* __restrict__ taus, const int* __restrict__ work_g,
    const float* __restrict__ background, const int* __restrict__ wptr,
    float* __restrict__ out, int P, int K)
{
    int p = blockIdx.x * TPB + threadIdx.x;
    if (p >= P) return;
    const int W   = wptr[0];
    const float x = (float)(p % W) + 0.5f;
    const float y = (float)(p / W) + 0.5f;

    float T = 1.f, cr = 0.f, cg = 0.f, cb = 0.f;
    const size_t base = (size_t)p * K;
    for (int k = 0; k < K; ++k) {
        int gi = work_g[base + k];
        float dx = x - means[2*gi];
        float dy = y - means[2*gi + 1];
        float a = conics[3*gi], b = conics[3*gi + 1], c = conics[3*gi + 2];
        float q  = fmaf(a * dx, dx, fmaf(2.f * b * dx, dy, c * dy * dy));
        float al = (q <= taus[gi]) ? opac[gi] * __expf(-0.5f * q) : 0.f;
        al = fminf(al, 0.99f);
        float w = al * T;
        cr = fmaf(w, colors[3*gi],     cr);
        cg = fmaf(w, colors[3*gi + 1], cg);
        cb = fmaf(w, colors[3*gi + 2], cb);
        T *= fmaxf(1.f - al, 1e-8f);
    }
    out[3*(size_t)p + 0] = fmaf(T, background[0], cr);
    out[3*(size_t)p + 1] = fmaf(T, background[1], cg);
    out[3*(size_t)p + 2] = fmaf(T, background[2], cb);
}

extern "C" void kernel_launch(void* const* d_in, const int* in_sizes, int n_in,
                              void* d_out, int out_size, void* d_ws, size_t ws_size,
                              hipStream_t stream)
{
    const float* means      = (const float*)d_in[0];
    const float* conics     = (const float*)d_in[1];
    const float* colors     = (const float*)d_in[2];
    const float* opac       = (const float*)d_in[3];
    const float* background = (const float*)d_in[4];
    const float* taus       = (const float*)d_in[5];
    const int*   work_g     = (const int*)d_in[6];
    const int*   wptr       = (const int*)d_in[13];   // image_width (device scalar)

    const int N = in_sizes[3];                        // opacities_s -> N_GAUSS
    const int M = in_sizes[6];                        // work_g      -> P*K
    const int P = in_sizes[11];                       // seg_end_idx -> num pixels
    const int K = (P > 0) ? (M / P) : 0;

    float* out = (float*)d_out;
    const size_t packedBytes = (size_t)N * 64;

    const bool fast = (K == KK) && (P % TPB == 0) && (ws_size >= packedBytes);
    if (fast) {
        float4* packed = (float4*)d_ws;
        pack_gauss<<<(N + TPB - 1) / TPB, TPB, 0, stream>>>(
            means, conics, colors, opac, taus, packed, N);
        render_k32<<<P / TPB, TPB, 0, stream>>>(
            packed, work_g, background, wptr, out);
    } else {
        render_generic<<<(P + TPB - 1) / TPB, TPB, 0, stream>>>(
            means, conics, colors, opac, taus, work_g, background, wptr, out, P, K);
    }
}